// Torch_Ops_Aten_GridSampler3DBackwardModule_53987738910952
// MI455X (gfx1250) — compile-verified
//
#include <hip/hip_runtime.h>
#include <hip/hip_bf16.h>

// Problem constants (fixed by setup_inputs)
#define NN 4
#define CC 32
#define DD 64
#define HH 64
#define WW 64
#define DO 32
#define HO 32
#define WO 32
#define SPATIAL_OUT (DO*HO*WO)              // 32768 points per batch
#define NPOINTS (NN*SPATIAL_OUT)            // 131072 total points
#define VOXELS (DD*HH*WW)                   // 262144 per (n,c)
#define GRADIN_ELEMS (NN*CC*VOXELS)         // 33554432

#if defined(__HIP_DEVICE_COMPILE__) && __has_builtin(__builtin_amdgcn_global_load_async_to_lds_b128)
#define ASYNC_LDS_OK 1
typedef int v4i_async __attribute__((vector_size(16)));
#else
#define ASYNC_LDS_OK 0
#endif

__global__ void gs3d_bwd_zero_kernel(float4* __restrict__ p) {
    int i = blockIdx.x * blockDim.x + threadIdx.x;   // exactly GRADIN_ELEMS/4 threads launched
    p[i] = make_float4(0.f, 0.f, 0.f, 0.f);
}

// One lane = one output grid point; loop over the 32 channels.
// Block = 256 threads = 256 consecutive points; grid slice staged via async LDS copy.
__global__ void __launch_bounds__(256)
gs3d_bwd_main_kernel(const float* __restrict__ grad_out,   // [N,C,Do,Ho,Wo]
                     const float* __restrict__ input,      // [N,C,D,H,W]
                     const float* __restrict__ grid,       // [N,Do,Ho,Wo,3]
                     float* __restrict__ grad_input,       // [N,C,D,H,W]
                     float* __restrict__ grad_grid) {      // [N,Do,Ho,Wo,3]
    __shared__ float sgrid[256 * 3];

    const int tid = threadIdx.x;
    const int p   = blockIdx.x * 256 + tid;

    // ---- Stage this block's 768 grid floats (3 KB) into LDS ----
    const float* gblk = grid + (size_t)blockIdx.x * (256 * 3);
    if (tid < 192) {  // 192 x 16B = 3072B
#if ASYNC_LDS_OK
        __builtin_amdgcn_global_load_async_to_lds_b128(
            (v4i_async*)(gblk + tid * 4), (v4i_async*)(&sgrid[tid * 4]),
            /*offset=*/0, /*cpol=*/0);
#else
        ((float4*)sgrid)[tid] = ((const float4*)gblk)[tid];
#endif
    }
#if ASYNC_LDS_OK
#if __has_builtin(__builtin_amdgcn_s_wait_asynccnt)
    __builtin_amdgcn_s_wait_asynccnt(0);
#else
    asm volatile("s_wait_asynccnt 0" ::: "memory");
#endif
#endif
    __syncthreads();

    const int n  = p >> 15;        // / SPATIAL_OUT
    const int sp = p & 32767;      // % SPATIAL_OUT

    // ---- Unnormalize (align_corners=1): ix = (g+1)*0.5*(dim-1), dim-1 = 63 ----
    const float gx = sgrid[3 * tid + 0];
    const float gy = sgrid[3 * tid + 1];
    const float gz = sgrid[3 * tid + 2];
    const float ix = (gx + 1.0f) * 31.5f;
    const float iy = (gy + 1.0f) * 31.5f;
    const float iz = (gz + 1.0f) * 31.5f;

    const float fx = floorf(ix), fy = floorf(iy), fz = floorf(iz);
    const float tx = ix - fx,    ty = iy - fy,    tz = iz - fz;
    const int x0 = (int)fx, y0 = (int)fy, z0 = (int)fz;
    const int x1 = x0 + 1,  y1 = y0 + 1,  z1 = z0 + 1;

    const bool bx0 = (x0 >= 0) & (x0 < WW), bx1 = (x1 >= 0) & (x1 < WW);
    const bool by0 = (y0 >= 0) & (y0 < HH), by1 = (y1 >= 0) & (y1 < HH);
    const bool bz0 = (z0 >= 0) & (z0 < DD), bz1 = (z1 >= 0) & (z1 < DD);
    const int cx0 = min(max(x0, 0), WW - 1), cx1 = min(max(x1, 0), WW - 1);
    const int cy0 = min(max(y0, 0), HH - 1), cy1 = min(max(y1, 0), HH - 1);
    const int cz0 = min(max(z0, 0), DD - 1), cz1 = min(max(z1, 0), DD - 1);

    // Voxel offsets inside one (n,c) volume
    const int oz0 = cz0 * (HH * WW), oz1 = cz1 * (HH * WW);
    const int oy0 = cy0 * WW,        oy1 = cy1 * WW;
    const int o000 = oz0 + oy0 + cx0, o100 = oz0 + oy0 + cx1;
    const int o010 = oz0 + oy1 + cx0, o110 = oz0 + oy1 + cx1;
    const int o001 = oz1 + oy0 + cx0, o101 = oz1 + oy0 + cx1;
    const int o011 = oz1 + oy1 + cx0, o111 = oz1 + oy1 + cx1;

    const bool m000 = bz0 & by0 & bx0, m100 = bz0 & by0 & bx1;
    const bool m010 = bz0 & by1 & bx0, m110 = bz0 & by1 & bx1;
    const bool m001 = bz1 & by0 & bx0, m101 = bz1 & by0 & bx1;
    const bool m011 = bz1 & by1 & bx0, m111 = bz1 & by1 & bx1;

    const float wx0 = 1.f - tx, wx1 = tx;
    const float wy0 = 1.f - ty, wy1 = ty;
    const float wz0 = 1.f - tz, wz1 = tz;

    // Trilinear weights (for grad_input scatter)
    const float w000 = wx0 * wy0 * wz0, w100 = wx1 * wy0 * wz0;
    const float w010 = wx0 * wy1 * wz0, w110 = wx1 * wy1 * wz0;
    const float w001 = wx0 * wy0 * wz1, w101 = wx1 * wy0 * wz1;
    const float w011 = wx0 * wy1 * wz1, w111 = wx1 * wy1 * wz1;

    // Pairwise products (for grad_grid)
    const float wy0z0 = wy0 * wz0, wy1z0 = wy1 * wz0, wy0z1 = wy0 * wz1, wy1z1 = wy1 * wz1;
    const float wx0z0 = wx0 * wz0, wx1z0 = wx1 * wz0, wx0z1 = wx0 * wz1, wx1z1 = wx1 * wz1;
    const float wx0y0 = wx0 * wy0, wx1y0 = wx1 * wy0, wx0y1 = wx0 * wy1, wx1y1 = wx1 * wy1;

    const float* gO_base = grad_out  + (size_t)n * (CC * SPATIAL_OUT) + sp;  // stride SPATIAL_OUT per c
    const float* in_base = input     + (size_t)n * (CC * VOXELS);
    float*       gi_base = grad_input + (size_t)n * (CC * VOXELS);

    float gix = 0.f, giy = 0.f, giz = 0.f;

#pragma unroll 2
    for (int c = 0; c < CC; ++c) {
        const float go = gO_base[c * SPATIAL_OUT];            // coalesced across lanes
        if (c + 1 < CC) __builtin_prefetch(gO_base + (c + 1) * SPATIAL_OUT, 0, 0);

        const float* b  = in_base + c * VOXELS;               // L2-resident gather
        float*       gb = gi_base + c * VOXELS;

        const float v000 = m000 ? b[o000] : 0.f;
        const float v100 = m100 ? b[o100] : 0.f;
        const float v010 = m010 ? b[o010] : 0.f;
        const float v110 = m110 ? b[o110] : 0.f;
        const float v001 = m001 ? b[o001] : 0.f;
        const float v101 = m101 ? b[o101] : 0.f;
        const float v011 = m011 ? b[o011] : 0.f;
        const float v111 = m111 ? b[o111] : 0.f;

        if (m000) atomicAdd(gb + o000, go * w000);
        if (m100) atomicAdd(gb + o100, go * w100);
        if (m010) atomicAdd(gb + o010, go * w010);
        if (m110) atomicAdd(gb + o110, go * w110);
        if (m001) atomicAdd(gb + o001, go * w001);
        if (m101) atomicAdd(gb + o101, go * w101);
        if (m011) atomicAdd(gb + o011, go * w011);
        if (m111) atomicAdd(gb + o111, go * w111);

        gix += go * ((v100 - v000) * wy0z0 + (v110 - v010) * wy1z0 +
                     (v101 - v001) * wy0z1 + (v111 - v011) * wy1z1);
        giy += go * ((v010 - v000) * wx0z0 + (v110 - v100) * wx1z0 +
                     (v011 - v001) * wx0z1 + (v111 - v101) * wx1z1);
        giz += go * ((v001 - v000) * wx0y0 + (v101 - v100) * wx1y0 +
                     (v011 - v010) * wx0y1 + (v111 - v110) * wx1y1);
    }

    // d ix / d gx = 0.5*(W-1) = 31.5 (same for all dims here)
    grad_grid[3 * p + 0] = gix * 31.5f;
    grad_grid[3 * p + 1] = giy * 31.5f;
    grad_grid[3 * p + 2] = giz * 31.5f;
}

extern "C" void kernel_launch(void* const* d_in, const int* in_sizes, int n_in,
                              void* d_out, int out_size, void* d_ws, size_t ws_size,
                              hipStream_t stream) {
    (void)in_sizes; (void)n_in; (void)out_size; (void)d_ws; (void)ws_size;

    const float* grad_out = (const float*)d_in[0];
    const float* input    = (const float*)d_in[1];
    const float* grid     = (const float*)d_in[2];
    // d_in[3..6] = interpolation_mode(0), padding_mode(0), align_corners(1), output_mask(3): fixed

    float* grad_input = (float*)d_out;
    float* grad_grid  = grad_input + GRADIN_ELEMS;

    // Zero grad_input (atomically accumulated); grad_grid is fully overwritten.
    {
        const int n4 = GRADIN_ELEMS / 4;              // 8388608 float4s
        gs3d_bwd_zero_kernel<<<n4 / 256, 256, 0, stream>>>((float4*)grad_input);
    }
    // Main backward: 131072 points, 256 points/block.
    gs3d_bwd_main_kernel<<<NPOINTS / 256, 256, 0, stream>>>(
        grad_out, input, grid, grad_input, grad_grid);
}